// VAELoss_76811195122191
// MI455X (gfx1250) — compile-verified
//
#include <hip/hip_runtime.h>
#include <hip/hip_bf16.h>
#include <math.h>

#define KL_EPS 1e-6f

typedef __attribute__((ext_vector_type(2))) float v2f;
typedef __attribute__((ext_vector_type(8))) float v8f;

// Per-wave LDS fence: DS ops are in-order within a wave on CDNA5; this stops
// compiler reordering/caching and drains DScnt for cross-lane visibility.
__device__ __forceinline__ void wave_lds_fence() {
    asm volatile("s_wait_dscnt 0" ::: "memory");
}

__device__ __forceinline__ float wave_reduce_sum(float v) {
#pragma unroll
    for (int off = 16; off > 0; off >>= 1) v += __shfl_xor(v, off, 32);
    return v;
}

// In-place lower Cholesky of a DIMxDIM SPD matrix in LDS (row stride S),
// executed by one wave32. Returns prod(diag(L)) (uniform across lanes).
template <int DIM, int S>
__device__ float wave_cholesky(float* M, int lane) {
    float prod = 1.0f;
    for (int k = 0; k < DIM; ++k) {
        float d = M[k * S + k];          // broadcast read
        float s = sqrtf(d);
        prod *= s;
        float inv = 1.0f / s;
        if (lane == 0) M[k * S + k] = s;
        for (int i = k + 1 + lane; i < DIM; i += 32) M[i * S + k] *= inv;
        wave_lds_fence();
        // rank-1 trailing update, lower triangle only
        for (int i = k + 1 + lane; i < DIM; i += 32) {
            float lik = M[i * S + k];
            for (int j = k + 1; j <= i; ++j) M[i * S + j] -= lik * M[j * S + k];
        }
        wave_lds_fence();
    }
    return prod;
}

// One wave (block of 32) per batch element.
// ws[b] = -0.5 * ( ||L_B^{-1} L_A||_F^2 - 64 + log((sdetA+eps)/(sdetB+eps)) )
__global__ __launch_bounds__(32) void kl_rec_kernel(const float* __restrict__ recon,
                                                    const float* __restrict__ orig,
                                                    float* __restrict__ ws, int nbatch) {
    constexpr int D = 64, S = 65;
    __shared__ float lds[2 * D * S];
    float* LB = lds;           // recon -> L_B
    float* LA = lds + D * S;   // orig  -> L_A -> X (in place)

    int b = blockIdx.x;
    if (b >= nbatch) return;
    int lane = threadIdx.x;

    const float4* gB = (const float4*)(recon + (size_t)b * D * D);
    const float4* gA = (const float4*)(orig + (size_t)b * D * D);
    for (int t = lane; t < D * D / 4; t += 32) {
        int r = t >> 4;            // 16 float4 per 64-wide row
        int c = (t & 15) * 4;
        float4 vb = gB[t];
        float4 va = gA[t];
        LB[r * S + c + 0] = vb.x; LB[r * S + c + 1] = vb.y;
        LB[r * S + c + 2] = vb.z; LB[r * S + c + 3] = vb.w;
        LA[r * S + c + 0] = va.x; LA[r * S + c + 1] = va.y;
        LA[r * S + c + 2] = va.z; LA[r * S + c + 3] = va.w;
    }
    wave_lds_fence();

    float prodB = wave_cholesky<D, S>(LB, lane);
    float prodA = wave_cholesky<D, S>(LA, lane);

    // Blocked TRSM: X = L_B^{-1} L_A, 16x16 tiles, lower blocks only.
    // Off-diagonal tile updates via V_WMMA_F32_16X16X4_F32.
    const int m = lane & 15;       // row/col-in-tile for A / B / C layouts
    const int kh = lane >> 4;      // lane half selects k-pairs / row-halves
    float fro = 0.0f;

    for (int J = 0; J < 4; ++J) {
        for (int I = J; I < 4; ++I) {
            v8f acc = {0.f, 0.f, 0.f, 0.f, 0.f, 0.f, 0.f, 0.f};
            for (int K = J; K < I; ++K) {
#pragma unroll
                for (int c4 = 0; c4 < 4; ++c4) {
                    // A slice: L_B[I,K], 16x4. ISA layout: vgpr v, lane-half kh -> k = v + 2*kh
                    int kidx = K * 16 + c4 * 4 + 2 * kh;
                    v2f a;
                    a.x = LB[(I * 16 + m) * S + kidx + 0];
                    a.y = LB[(I * 16 + m) * S + kidx + 1];
                    // B slice: X[K,J], 4x16, same k mapping, col = lane&15
                    v2f bb;
                    bb.x = LA[(kidx + 0) * S + J * 16 + m];
                    bb.y = LA[(kidx + 1) * S + J * 16 + m];
                    acc = __builtin_amdgcn_wmma_f32_16x16x4_f32(
                        false, a, false, bb, (short)0, acc, false, false);
                }
            }
            // T = L_A[I,J] - acc ; zero strict upper of diagonal blocks
            // C/D layout: vgpr v -> row v + 8*kh, col = lane&15
#pragma unroll
            for (int v = 0; v < 8; ++v) {
                int r = v + 8 * kh;
                float t = LA[(I * 16 + r) * S + J * 16 + m] - acc[v];
                if (I == J && r < m) t = 0.0f;
                LA[(I * 16 + r) * S + J * 16 + m] = t;
            }
            wave_lds_fence();
            // 16x16 triangular solve L_B[I,I] * Xblk = T: one column per lane (0..15)
            if (lane < 16) {
                float tc[16];
#pragma unroll
                for (int r = 0; r < 16; ++r) tc[r] = LA[(I * 16 + r) * S + J * 16 + lane];
                for (int r = 0; r < 16; ++r) {
                    float x = tc[r] / LB[(I * 16 + r) * S + I * 16 + r];
                    tc[r] = x;
                    for (int rr = r + 1; rr < 16; ++rr)
                        tc[rr] -= LB[(I * 16 + rr) * S + I * 16 + r] * x;
                }
#pragma unroll
                for (int r = 0; r < 16; ++r) {
                    float x = tc[r];
                    if (I == J && r < lane) x = 0.0f;
                    LA[(I * 16 + r) * S + J * 16 + lane] = x;
                    fro += x * x;
                }
            }
            wave_lds_fence();
        }
    }

    float tr = wave_reduce_sum(fro);   // = tr(B^{-1} A)
    if (lane == 0) {
        float logdet = logf((prodA + KL_EPS) / (prodB + KL_EPS));
        float kl = 0.5f * (tr - (float)D + logdet);
        ws[b] = -kl;
    }
}

// KL(encoded || I): 4 waves per block, one 32x32 matrix per wave.
__global__ __launch_bounds__(128) void kl_enc_kernel(const float* __restrict__ enc,
                                                     float* __restrict__ ws, int nbatch) {
    constexpr int D = 32, S = 33, WPB = 4;
    __shared__ float lds[WPB * D * S];
    int wave = threadIdx.x >> 5;
    int lane = threadIdx.x & 31;
    int b = blockIdx.x * WPB + wave;
    if (b >= nbatch) return;
    float* M = lds + wave * D * S;

    const float4* g = (const float4*)(enc + (size_t)b * D * D);
    for (int t = lane; t < D * D / 4; t += 32) {
        int r = t >> 3;            // 8 float4 per 32-wide row
        int c = (t & 7) * 4;
        float4 v = g[t];
        M[r * S + c + 0] = v.x; M[r * S + c + 1] = v.y;
        M[r * S + c + 2] = v.z; M[r * S + c + 3] = v.w;
    }
    wave_lds_fence();

    float tr = wave_reduce_sum(M[lane * S + lane]);   // trace of raw matrix
    float prod = wave_cholesky<D, S>(M, lane);
    if (lane == 0) {
        float kl = 0.5f * (tr - (float)D + logf((prod + KL_EPS) / (1.0f + KL_EPS)));
        ws[b] = -kl;
    }
}

// Deterministic final reduction: ws[0..n) = per-batch -kl_rec, ws[n..2n) = -kl_enc.
__global__ __launch_bounds__(256) void kl_reduce_kernel(const float* __restrict__ ws, int n,
                                                        float* __restrict__ out) {
    __shared__ float sr[256];
    __shared__ float se[256];
    int tid = threadIdx.x;
    float a = 0.0f, e = 0.0f;
    for (int i = tid; i < n; i += 256) { a += ws[i]; e += ws[n + i]; }
    sr[tid] = a; se[tid] = e;
    __syncthreads();
    for (int s = 128; s > 0; s >>= 1) {
        if (tid < s) { sr[tid] += sr[tid + s]; se[tid] += se[tid + s]; }
        __syncthreads();
    }
    if (tid == 0) {
        out[0] = sr[0] + se[0];   // loss
        out[1] = sr[0];           // kl_loss_reconstruction
        out[2] = se[0];           // kl_loss_encoding
    }
}

extern "C" void kernel_launch(void* const* d_in, const int* in_sizes, int n_in,
                              void* d_out, int out_size, void* d_ws, size_t ws_size,
                              hipStream_t stream) {
    const float* recon = (const float*)d_in[0];   // [B,64,64]
    const float* orig  = (const float*)d_in[1];   // [B,64,64]
    const float* enc   = (const float*)d_in[2];   // [B,32,32]
    float* out = (float*)d_out;
    float* ws  = (float*)d_ws;

    const int B  = in_sizes[0] / (64 * 64);
    const int Be = in_sizes[2] / (32 * 32);       // == B for this problem

    // one wave32 per 64x64 pair
    kl_rec_kernel<<<B, 32, 0, stream>>>(recon, orig, ws, B);
    // 4 matrices per 128-thread block
    kl_enc_kernel<<<(Be + 3) / 4, 128, 0, stream>>>(enc, ws + B, Be);
    // deterministic scalar reduction
    kl_reduce_kernel<<<1, 256, 0, stream>>>(ws, B, out);
    (void)n_in; (void)out_size; (void)ws_size;
}